// HeteroGNN_11682311045362
// MI455X (gfx1250) — compile-verified
//
#include <hip/hip_runtime.h>

typedef __attribute__((ext_vector_type(2))) float v2f;
typedef __attribute__((ext_vector_type(8))) float v8f;

#define C_DIM 128

// ---------------------------------------------------------------------------
// Zero-fill scratch (msum + cnt region) before atomic accumulation.
// ---------------------------------------------------------------------------
__global__ void __launch_bounds__(256) fill0_kernel(float* __restrict__ p, size_t n) {
    size_t i = (size_t)blockIdx.x * blockDim.x + threadIdx.x;
    if (i < n) p[i] = 0.0f;
}

// ---------------------------------------------------------------------------
// Scatter-add: one wave (32 lanes) per edge. Each lane handles 4 channels:
// coalesced float4 gather of the source row (L2-resident) + 4 f32 atomics
// into msum[dst]. Lane 0 bumps the degree counter.
// ---------------------------------------------------------------------------
__global__ void __launch_bounds__(256) scatter_mean_kernel(
    const float* __restrict__ xsrc, const int* __restrict__ src,
    const int* __restrict__ dst, float* __restrict__ msum,
    float* __restrict__ cnt, int n_edges) {
    int gid  = blockIdx.x * blockDim.x + threadIdx.x;
    int e    = gid >> 5;
    int lane = threadIdx.x & 31;
    if (e >= n_edges) return;
    int s = src[e];
    int d = dst[e];
    const float4 v = *(const float4*)(xsrc + (size_t)s * C_DIM + lane * 4);
    float* o = msum + (size_t)d * C_DIM + lane * 4;
    atomicAdd(o + 0, v.x);
    atomicAdd(o + 1, v.y);
    atomicAdd(o + 2, v.z);
    atomicAdd(o + 3, v.w);
    if (lane == 0) atomicAdd(cnt + d, 1.0f);
}

// ---------------------------------------------------------------------------
// One K=128 GEMM slab on the fp32 WMMA pipe: acc += (scale*Arow) x Wrow.
// A 16x4 fp32 layout: lane l -> m=l%16, k-pair at koff=2*(l/16) (contig f2).
// B  4x16 fp32 layout: lane l -> n=l%16, B[k][n]=W[n][k] (contig f2 in W row).
// Arow/Wrow are pre-offset by koff by the caller.
// ---------------------------------------------------------------------------
__device__ __forceinline__ v8f wmma_gemm128(const float* __restrict__ Arow,
                                            float scale,
                                            const float* __restrict__ Wrow,
                                            v8f acc) {
#pragma unroll
    for (int k0 = 0; k0 < C_DIM; k0 += 4) {
        v2f a = *(const v2f*)(Arow + k0);
        a.x *= scale;
        a.y *= scale;
        v2f b = *(const v2f*)(Wrow + k0);
        acc = __builtin_amdgcn_wmma_f32_16x16x4_f32(
            /*neg_a=*/false, a, /*neg_b=*/false, b,
            /*c_mod=*/(short)0, acc, /*reuse_a=*/false, /*reuse_b=*/false);
    }
    return acc;
}

// ---------------------------------------------------------------------------
// Fused SAGE output kernel: out = relu?( mean1@Wl1^T + xdst@Wr1^T + bl1
//                                       [+ mean2@Wl2^T + xdst@Wr2^T + bl2] )
// Block = 256 threads = 8 waves; blockIdx.x = 16-row M tile, wave id = N tile
// (8*16 = 128 = C). EXEC is all-1s everywhere (WMMA requirement).
// Two accumulator chains (mean-GEMM / x-GEMM) for WMMA ILP.
// ---------------------------------------------------------------------------
__global__ void __launch_bounds__(256) sage_wmma_kernel(
    const float* __restrict__ msum1, const float* __restrict__ cnt1,
    const float* __restrict__ Wl1, const float* __restrict__ bl1,
    const float* __restrict__ Wr1,
    const float* __restrict__ msum2, const float* __restrict__ cnt2,
    const float* __restrict__ Wl2, const float* __restrict__ bl2,
    const float* __restrict__ Wr2,
    const float* __restrict__ xdst, float* __restrict__ out,
    int do_relu, int has2) {
    const int mt   = blockIdx.x;
    const int nt   = threadIdx.x >> 5;
    const int lane = threadIdx.x & 31;
    const int half = lane >> 4;
    const int l16  = lane & 15;
    const int m    = mt * 16 + l16;   // A row for this lane
    const int n    = nt * 16 + l16;   // B/D column for this lane
    const int koff = half * 2;

    const float* ax = xdst + (size_t)m * C_DIM + koff;

    v8f accA = {};   // mean @ Wl chains
    v8f accB = {};   // x_dst @ Wr chains

    float inv1 = 1.0f / fmaxf(cnt1[m], 1.0f);
    accA = wmma_gemm128(msum1 + (size_t)m * C_DIM + koff, inv1,
                        Wl1 + (size_t)n * C_DIM + koff, accA);
    accB = wmma_gemm128(ax, 1.0f, Wr1 + (size_t)n * C_DIM + koff, accB);
    float bias = bl1[n];

    if (has2) {
        float inv2 = 1.0f / fmaxf(cnt2[m], 1.0f);
        accA = wmma_gemm128(msum2 + (size_t)m * C_DIM + koff, inv2,
                            Wl2 + (size_t)n * C_DIM + koff, accA);
        accB = wmma_gemm128(ax, 1.0f, Wr2 + (size_t)n * C_DIM + koff, accB);
        bias += bl2[n];
    }

    // C/D 16x16 f32 layout: VGPR j -> row m = mt*16 + j + 8*half, col = n.
#pragma unroll
    for (int j = 0; j < 8; ++j) {
        float v = accA[j] + accB[j] + bias;
        if (do_relu) v = fmaxf(v, 0.0f);
        out[(size_t)(mt * 16 + j + half * 8) * C_DIM + n] = v;
    }
}

// ---------------------------------------------------------------------------
// Launcher
// ---------------------------------------------------------------------------
extern "C" void kernel_launch(void* const* d_in, const int* in_sizes, int n_in,
                              void* d_out, int out_size, void* d_ws, size_t ws_size,
                              hipStream_t stream) {
    const int* src_g2t = (const int*)d_in[0];
    const int* dst_g2t = (const int*)d_in[1];
    const int* src_t2g = (const int*)d_in[2];
    const int* dst_t2g = (const int*)d_in[3];
    const int* src_g2g = (const int*)d_in[4];
    const int* dst_g2g = (const int*)d_in[5];
    const float* emb_gene  = (const float*)d_in[6];
    const float* emb_trait = (const float*)d_in[7];

    const int E_GT = in_sizes[0];
    const int E_TG = in_sizes[2];
    const int E_GG = in_sizes[4];
    const int N_GENE  = in_sizes[6] / C_DIM;
    const int N_TRAIT = in_sizes[7] / C_DIM;

    // weights: [8..16] layer1, [17..25] layer2; per edge-type: Wl, bl, Wr
    const float* l1_g2t_Wl = (const float*)d_in[8];
    const float* l1_g2t_bl = (const float*)d_in[9];
    const float* l1_g2t_Wr = (const float*)d_in[10];
    const float* l1_t2g_Wl = (const float*)d_in[11];
    const float* l1_t2g_bl = (const float*)d_in[12];
    const float* l1_t2g_Wr = (const float*)d_in[13];
    const float* l1_g2g_Wl = (const float*)d_in[14];
    const float* l1_g2g_bl = (const float*)d_in[15];
    const float* l1_g2g_Wr = (const float*)d_in[16];
    const float* l2_g2t_Wl = (const float*)d_in[17];
    const float* l2_g2t_bl = (const float*)d_in[18];
    const float* l2_g2t_Wr = (const float*)d_in[19];
    const float* l2_t2g_Wl = (const float*)d_in[20];
    const float* l2_t2g_bl = (const float*)d_in[21];
    const float* l2_t2g_Wr = (const float*)d_in[22];
    const float* l2_g2g_Wl = (const float*)d_in[23];
    const float* l2_g2g_bl = (const float*)d_in[24];
    const float* l2_g2g_Wr = (const float*)d_in[25];

    // Workspace layout (floats)
    float* ws = (float*)d_ws;
    size_t off = 0;
    float* msumT  = ws + off; off += (size_t)N_TRAIT * C_DIM;
    float* cntT   = ws + off; off += (size_t)N_TRAIT;
    float* msumG1 = ws + off; off += (size_t)N_GENE * C_DIM;
    float* cntG1  = ws + off; off += (size_t)N_GENE;
    float* msumG2 = ws + off; off += (size_t)N_GENE * C_DIM;
    float* cntG2  = ws + off; off += (size_t)N_GENE;
    const size_t zero_n = off;            // atomics region, re-zeroed per layer
    float* zg = ws + off; off += (size_t)N_GENE * C_DIM;
    float* zt = ws + off; off += (size_t)N_TRAIT * C_DIM;

    float* out_gene  = (float*)d_out;
    float* out_trait = out_gene + (size_t)N_GENE * C_DIM;

    const int TPB = 256;
    dim3 blk(TPB);
    unsigned gz  = (unsigned)((zero_n + TPB - 1) / TPB);
    unsigned ggt = (unsigned)(((size_t)E_GT * 32 + TPB - 1) / TPB);
    unsigned gtg = (unsigned)(((size_t)E_TG * 32 + TPB - 1) / TPB);
    unsigned ggg = (unsigned)(((size_t)E_GG * 32 + TPB - 1) / TPB);
    unsigned gmT = (unsigned)(N_TRAIT / 16);
    unsigned gmG = (unsigned)(N_GENE / 16);

    // ---------------- Layer 1 ----------------
    fill0_kernel<<<gz, blk, 0, stream>>>(ws, zero_n);
    scatter_mean_kernel<<<ggt, blk, 0, stream>>>(emb_gene,  src_g2t, dst_g2t, msumT,  cntT,  E_GT);
    scatter_mean_kernel<<<gtg, blk, 0, stream>>>(emb_trait, src_t2g, dst_t2g, msumG1, cntG1, E_TG);
    scatter_mean_kernel<<<ggg, blk, 0, stream>>>(emb_gene,  src_g2g, dst_g2g, msumG2, cntG2, E_GG);
    sage_wmma_kernel<<<gmT, blk, 0, stream>>>(
        msumT, cntT, l1_g2t_Wl, l1_g2t_bl, l1_g2t_Wr,
        nullptr, nullptr, nullptr, nullptr, nullptr,
        emb_trait, zt, /*relu=*/1, /*has2=*/0);
    sage_wmma_kernel<<<gmG, blk, 0, stream>>>(
        msumG1, cntG1, l1_t2g_Wl, l1_t2g_bl, l1_t2g_Wr,
        msumG2, cntG2, l1_g2g_Wl, l1_g2g_bl, l1_g2g_Wr,
        emb_gene, zg, /*relu=*/1, /*has2=*/1);

    // ---------------- Layer 2 ----------------
    fill0_kernel<<<gz, blk, 0, stream>>>(ws, zero_n);
    scatter_mean_kernel<<<ggt, blk, 0, stream>>>(zg, src_g2t, dst_g2t, msumT,  cntT,  E_GT);
    scatter_mean_kernel<<<gtg, blk, 0, stream>>>(zt, src_t2g, dst_t2g, msumG1, cntG1, E_TG);
    scatter_mean_kernel<<<ggg, blk, 0, stream>>>(zg, src_g2g, dst_g2g, msumG2, cntG2, E_GG);
    sage_wmma_kernel<<<gmT, blk, 0, stream>>>(
        msumT, cntT, l2_g2t_Wl, l2_g2t_bl, l2_g2t_Wr,
        nullptr, nullptr, nullptr, nullptr, nullptr,
        zt, out_trait, /*relu=*/0, /*has2=*/0);
    sage_wmma_kernel<<<gmG, blk, 0, stream>>>(
        msumG1, cntG1, l2_t2g_Wl, l2_t2g_bl, l2_t2g_Wr,
        msumG2, cntG2, l2_g2g_Wl, l2_g2g_bl, l2_g2g_Wr,
        zg, out_gene, /*relu=*/0, /*has2=*/1);

    (void)n_in; (void)out_size; (void)ws_size;
}